// Involution3d_51187420233902
// MI455X (gfx1250) — compile-verified
//
#include <hip/hip_runtime.h>

// ---------------- problem constants ----------------
#define CIN    8
#define COUT   16
#define NGRP   4
#define CMID   4
#define KNUM   343          // 7*7*7
#define DDIM   24
#define NVOX   13824        // 24^3
#define GRP_ROWS 352        // per-group rows padded to 22*16
#define NTILE  22           // 352/16 WMMA row-tiles per group
#define BN_EPS 1e-5f

typedef float v2f __attribute__((ext_vector_type(2)));
typedef float v8f __attribute__((ext_vector_type(8)));

// ---------------- k0: zero BN stats accumulators ----------------
__global__ void k0_zero(float* stats) {
    if (threadIdx.x < 8) stats[threadIdx.x] = 0.f;
}

// ---------------- k1: x_map (voxel-major), red (voxel-major), BN partial sums ----
__global__ void __launch_bounds__(256)
k1_map(const float* __restrict__ in, const float* __restrict__ w_init,
       const float* __restrict__ w_red, float* __restrict__ xmap,
       float* __restrict__ red, float* __restrict__ stats) {
    __shared__ float sWi[COUT * CIN];   // 128
    __shared__ float sWr[CMID * CIN];   // 32
    const int t = threadIdx.x;
    if (t < COUT * CIN) sWi[t] = w_init[t];
    if (t < CMID * CIN) sWr[t] = w_red[t];
    __syncthreads();

    const int p = blockIdx.x * 256 + t;      // grid sized exactly: 54*256 == NVOX
    float x[CIN];
#pragma unroll
    for (int i = 0; i < CIN; ++i) x[i] = in[i * NVOX + p];

#pragma unroll
    for (int c = 0; c < COUT; ++c) {
        float acc = 0.f;
#pragma unroll
        for (int i = 0; i < CIN; ++i) acc += sWi[c * CIN + i] * x[i];
        xmap[p * COUT + c] = acc;            // voxel-major: group => contiguous float4
    }

    float r[CMID], s[CMID], ss[CMID];
#pragma unroll
    for (int m = 0; m < CMID; ++m) {
        float acc = 0.f;
#pragma unroll
        for (int i = 0; i < CIN; ++i) acc += sWr[m * CIN + i] * x[i];
        r[m] = acc;
        red[p * CMID + m] = acc;             // voxel-major: one float4 per voxel
        s[m] = acc;
        ss[m] = acc * acc;
    }
    // wave32 tree reduction, one atomic per wave leader
#pragma unroll
    for (int off = 16; off > 0; off >>= 1) {
#pragma unroll
        for (int m = 0; m < CMID; ++m) {
            s[m]  += __shfl_down(s[m],  off, 32);
            ss[m] += __shfl_down(ss[m], off, 32);
        }
    }
    if ((t & 31) == 0) {
#pragma unroll
        for (int m = 0; m < CMID; ++m) {
            atomicAdd(&stats[m],     s[m]);
            atomicAdd(&stats[4 + m], ss[m]);
        }
    }
}

// ---------------- k2: fold BN into affine sig = relu(a*red + b) ----------------
__global__ void k2_bn(const float* __restrict__ stats, const float* __restrict__ gamma,
                      const float* __restrict__ beta, float* __restrict__ ab) {
    const int m = threadIdx.x;
    if (m < CMID) {
        const float inv  = 1.0f / (float)NVOX;
        const float mean = stats[m] * inv;
        const float var  = stats[4 + m] * inv - mean * mean;   // biased, torch BN training
        const float a    = gamma[m] * rsqrtf(var + BN_EPS);
        ab[m]     = a;
        ab[4 + m] = beta[m] - mean * a;
    }
}

// ---------------- k3: fused ker-GEMM (WMMA f32 16x16x4) + involution gather ------
__global__ void __launch_bounds__(256)
k3_main(const float* __restrict__ xmap, const float* __restrict__ red,
        const float* __restrict__ ab, const float* __restrict__ w_span,
        const float* __restrict__ b_span, float* __restrict__ out) {
    // w_span / b_span staged in LDS, each group padded 343 -> 352 rows with
    // zeros: pad rows yield ker == 0, so the hot loop needs no row guard and
    // every 16-row WMMA tile sits inside one group (compile-time g).
    __shared__ float sA[NGRP * GRP_ROWS * 4];   // 22528 B
    __shared__ float sBias[NGRP * GRP_ROWS];    //  5632 B
    __shared__ int   sBnd[GRP_ROWS];            //  1408 B  packed (dz,dy,dx)
    __shared__ int   sDelta[GRP_ROWS];          //  1408 B  linear voxel delta

    for (int i = threadIdx.x; i < NGRP * GRP_ROWS * 4; i += 256) {
        const int gi = i / (GRP_ROWS * 4);
        const int r  = (i - gi * GRP_ROWS * 4) >> 2;
        const int c  = i & 3;
        sA[i] = (r < KNUM) ? w_span[(gi * KNUM + r) * 4 + c] : 0.f;
    }
    for (int i = threadIdx.x; i < NGRP * GRP_ROWS; i += 256) {
        const int gi = i / GRP_ROWS;
        const int r  = i - gi * GRP_ROWS;
        sBias[i] = (r < KNUM) ? b_span[gi * KNUM + r] : 0.f;
    }
    for (int k = threadIdx.x; k < GRP_ROWS; k += 256) {
        int pk = 0, dl = 0;
        if (k < KNUM) {
            const int dz = k / 49;
            const int kr = k - dz * 49;
            const int dy = kr / 7;
            const int dx = kr - dy * 7;
            pk = dz | (dy << 8) | (dx << 16);
            dl = (dz - 3) * (DDIM * DDIM) + (dy - 3) * DDIM + (dx - 3);
        }
        sBnd[k] = pk;      // pad rows: (0,0,0)/0 -> harmless, ker==0 kills term
        sDelta[k] = dl;
    }
    __syncthreads();

    const int lane = threadIdx.x & 31;
    const int wave = threadIdx.x >> 5;
    const int tile = blockIdx.x * 8 + wave;      // 864 tiles of 16 voxels
    const int col  = lane & 15;
    const int hi   = lane >> 4;                  // half-wave select
    const int p    = tile * 16 + col;

    // voxel coords of this lane's column
    const int z  = p / (DDIM * DDIM);
    const int rm = p - z * DDIM * DDIM;
    const int y  = rm / DDIM;
    const int x  = rm - y * DDIM;

    // sig for this voxel (BN folded to affine, then ReLU)
    const float4 r4 = *(const float4*)(red + p * 4);
    const float sg0 = fmaxf(ab[0] * r4.x + ab[4], 0.f);
    const float sg1 = fmaxf(ab[1] * r4.y + ab[5], 0.f);
    const float sg2 = fmaxf(ab[2] * r4.z + ab[6], 0.f);
    const float sg3 = fmaxf(ab[3] * r4.w + ab[7], 0.f);
    // B matrix (4x16): low half-wave carries K=0/1, high half-wave K=2/3, N = col
    v2f Bm;
    Bm.x = hi ? sg2 : sg0;
    Bm.y = hi ? sg3 : sg1;

    float4 acc[NGRP];
#pragma unroll
    for (int g = 0; g < NGRP; ++g) { acc[g].x = acc[g].y = acc[g].z = acc[g].w = 0.f; }

    for (int kt = 0; kt < NTILE; ++kt) {
        const int rbase = kt * 16 + hi * 8;

        // ---- per-kt row metadata, shared by all 4 groups (branchless) ----
        const int4 b0 = *(const int4*)(sBnd + rbase);
        const int4 b1 = *(const int4*)(sBnd + rbase + 4);
        const int4 d0 = *(const int4*)(sDelta + rbase);
        const int4 d1 = *(const int4*)(sDelta + rbase + 4);
        const int bnd[8] = { b0.x, b0.y, b0.z, b0.w, b1.x, b1.y, b1.z, b1.w };
        const int dlt[8] = { d0.x, d0.y, d0.z, d0.w, d1.x, d1.y, d1.z, d1.w };
        unsigned msk[8];
        int      nps[8];
#pragma unroll
        for (int i = 0; i < 8; ++i) {
            const int nz = z + (bnd[i] & 0xff) - 3;
            const int ny = y + ((bnd[i] >> 8) & 0xff) - 3;
            const int nx = x + ((bnd[i] >> 16) & 0xff) - 3;
            const bool ok = ((unsigned)nz < DDIM) & ((unsigned)ny < DDIM) &
                            ((unsigned)nx < DDIM);
            msk[i] = ok ? 0xffffffffu : 0u;   // masks D element (0*xm exact)
            nps[i] = ok ? (p + dlt[i]) : p;   // clamp OOB to a safe address
        }

        // ---- 4 groups: WMMA ker-tile, then 8 batched gathers + masked FMAs ----
#pragma unroll
        for (int g = 0; g < NGRP; ++g) {
            const float* sAg = sA + g * GRP_ROWS * 4;
            const float* sBg = sBias + g * GRP_ROWS;
            // A tile: lane holds (M = col, K = 2*hi .. 2*hi+1)
            const v2f Am = *(const v2f*)(sAg + (kt * 16 + col) * 4 + hi * 2);
            // C = bias broadcast: VGPR i <-> row M = i + 8*hi
            const float4 cl0 = *(const float4*)(sBg + rbase);
            const float4 cl1 = *(const float4*)(sBg + rbase + 4);
            v8f Cm;
            Cm[0] = cl0.x; Cm[1] = cl0.y; Cm[2] = cl0.z; Cm[3] = cl0.w;
            Cm[4] = cl1.x; Cm[5] = cl1.y; Cm[6] = cl1.z; Cm[7] = cl1.w;

            // ker tile = w_span_tile @ sig_tile + bias  (V_WMMA_F32_16X16X4_F32)
            v8f Dm = __builtin_amdgcn_wmma_f32_16x16x4_f32(
                false, Am, false, Bm, (short)0, Cm, false, false);

            // 8 independent global_load_b128: issue all, then consume
            float4 xm[8];
#pragma unroll
            for (int i = 0; i < 8; ++i)
                xm[i] = *(const float4*)(xmap + nps[i] * COUT + g * 4);
#pragma unroll
            for (int i = 0; i < 8; ++i) {
                const float dv = __int_as_float(__float_as_int(Dm[i]) & msk[i]);
                acc[g].x += dv * xm[i].x; acc[g].y += dv * xm[i].y;
                acc[g].z += dv * xm[i].z; acc[g].w += dv * xm[i].w;
            }
        }
    }

    // lanes l and l+16 hold partial sums for the same column p (rows 0-7 vs 8-15)
#pragma unroll
    for (int g = 0; g < NGRP; ++g) {
        acc[g].x += __shfl_xor(acc[g].x, 16, 32);
        acc[g].y += __shfl_xor(acc[g].y, 16, 32);
        acc[g].z += __shfl_xor(acc[g].z, 16, 32);
        acc[g].w += __shfl_xor(acc[g].w, 16, 32);
    }

    if (hi == 0) {   // channel-major output [16][13824]
#pragma unroll
        for (int g = 0; g < NGRP; ++g) {
            out[(g * 4 + 0) * NVOX + p] = acc[g].x;
            out[(g * 4 + 1) * NVOX + p] = acc[g].y;
            out[(g * 4 + 2) * NVOX + p] = acc[g].z;
            out[(g * 4 + 3) * NVOX + p] = acc[g].w;
        }
    }
}

// ---------------- launch ----------------
extern "C" void kernel_launch(void* const* d_in, const int* in_sizes, int n_in,
                              void* d_out, int out_size, void* d_ws, size_t ws_size,
                              hipStream_t stream) {
    const float* in     = (const float*)d_in[0];
    const float* w_init = (const float*)d_in[1];
    const float* w_red  = (const float*)d_in[2];
    const float* gamma  = (const float*)d_in[3];
    const float* beta   = (const float*)d_in[4];
    const float* w_span = (const float*)d_in[5];
    const float* b_span = (const float*)d_in[6];
    float* out = (float*)d_out;

    char* ws = (char*)d_ws;
    float* stats = (float*)ws;                       // 8 floats: sum[4], sumsq[4]
    float* ab    = stats + 8;                        // 8 floats: a[4], b[4]
    float* red   = (float*)(ws + 256);               // [NVOX][4]  (221184 B)
    float* xmap  = (float*)(ws + 256 + (size_t)NVOX * CMID * sizeof(float)); // [NVOX][16]

    k0_zero<<<1, 32, 0, stream>>>(stats);
    k1_map<<<NVOX / 256, 256, 0, stream>>>(in, w_init, w_red, xmap, red, stats);
    k2_bn<<<1, 32, 0, stream>>>(stats, gamma, beta, ab);
    k3_main<<<(NVOX / 16) / 8, 256, 0, stream>>>(xmap, red, ab, w_span, b_span, out);
}